// Dist_11630771438032
// MI455X (gfx1250) — compile-verified
//
#include <hip/hip_runtime.h>

// CDNA5 / gfx1250: wave32, WMMA f32 16x16x4.
typedef __attribute__((ext_vector_type(2))) float v2f;
typedef __attribute__((ext_vector_type(8))) float v8f;

#define TILE      128
#define DDIM      128
#define LDS_STR   132            // 128 + 4 pad: conflict-free ds_load_b64 frags
#define NDIM      8192
#define MDIM      8192

__global__ __launch_bounds__(256, 2)
void dist_wmma_f32_kernel(const float* __restrict__ x,
                          const float* __restrict__ y,
                          float* __restrict__ out) {
    __shared__ float xs[TILE * LDS_STR];   // x tile, row-major [row][d]
    __shared__ float ys[TILE * LDS_STR];   // y tile, row-major [col][d]  (== B^T)
    __shared__ float xsq[TILE];
    __shared__ float ysq[TILE];

    const int tid     = threadIdx.x;
    const int lane    = tid & 31;
    const int wave    = tid >> 5;          // 0..7: wave owns rows 16w..16w+15
    const int lmod    = lane & 15;
    const int lhalf   = lane >> 4;         // 0 or 1
    const int rowBase = blockIdx.y * TILE; // n
    const int colBase = blockIdx.x * TILE; // m

    // ---- Global -> LDS tile copy (coalesced b128) ----
    {
        const int c4 = tid & 31;           // float4 column 0..31 (covers 128 floats)
        const int r0 = tid >> 5;           // 0..7
        #pragma unroll
        for (int i = 0; i < 16; ++i) {
            const int row = i * 8 + r0;
            const float4 vx = ((const float4*)(x + (size_t)(rowBase + row) * DDIM))[c4];
            *(float4*)&xs[row * LDS_STR + c4 * 4] = vx;
            const float4 vy = ((const float4*)(y + (size_t)(colBase + row) * DDIM))[c4];
            *(float4*)&ys[row * LDS_STR + c4 * 4] = vy;
        }
    }
    __syncthreads();

    // ---- Squared norms from LDS (one row per thread) ----
    {
        const float* r = (tid < TILE) ? &xs[tid * LDS_STR]
                                      : &ys[(tid - TILE) * LDS_STR];
        float s = 0.0f;
        #pragma unroll
        for (int d4 = 0; d4 < DDIM / 4; ++d4) {
            const float4 v = *(const float4*)(r + d4 * 4);
            s = fmaf(v.x, v.x, s);
            s = fmaf(v.y, v.y, s);
            s = fmaf(v.z, v.z, s);
            s = fmaf(v.w, v.w, s);
        }
        if (tid < TILE) xsq[tid] = s; else ysq[tid - TILE] = s;
    }
    __syncthreads();

    // ---- Main WMMA loop: wave computes 16 x 128 of the tile ----
    v8f acc[8];
    #pragma unroll
    for (int t = 0; t < 8; ++t) acc[t] = (v8f){0,0,0,0,0,0,0,0};

    // A frag: lane<16 -> (K0,K1) of row M=lmod; lane>=16 -> (K2,K3). One ds_load_b64.
    const float* arow  = &xs[(wave * 16 + lmod) * LDS_STR + 2 * lhalf];
    // B frag: lane<16 -> (K0,K1) of col N=lmod; lane>=16 -> (K2,K3). Same shape.
    const float* bbase = &ys[lmod * LDS_STR + 2 * lhalf];

    #pragma unroll 4
    for (int s = 0; s < DDIM / 4; ++s) {            // 32 K-steps of 4
        const v2f a = *(const v2f*)(arow + 4 * s);
        #pragma unroll
        for (int t = 0; t < 8; ++t) {
            const v2f b = *(const v2f*)(bbase + t * 16 * LDS_STR + 4 * s);
            acc[t] = __builtin_amdgcn_wmma_f32_16x16x4_f32(
                /*neg_a=*/false, a, /*neg_b=*/false, b,
                /*c_mod=*/(short)0, acc[t],
                /*reuse_a=*/false, /*reuse_b=*/false);
        }
    }

    // ---- Epilogue: dist = xsq + ysq - 2*cross, fused store ----
    float xq[8];
    #pragma unroll
    for (int v = 0; v < 8; ++v) xq[v] = xsq[wave * 16 + 8 * lhalf + v];

    #pragma unroll
    for (int t = 0; t < 8; ++t) {
        const float yq = ysq[t * 16 + lmod];
        #pragma unroll
        for (int v = 0; v < 8; ++v) {
            const int gr = rowBase + wave * 16 + 8 * lhalf + v;  // n
            const int gc = colBase + t * 16 + lmod;              // m
            out[(size_t)gr * MDIM + gc] = fmaf(-2.0f, acc[t][v], xq[v] + yq);
        }
    }
}

extern "C" void kernel_launch(void* const* d_in, const int* in_sizes, int n_in,
                              void* d_out, int out_size, void* d_ws, size_t ws_size,
                              hipStream_t stream) {
    const float* x = (const float*)d_in[0];   // [8192, 128] f32
    const float* y = (const float*)d_in[1];   // [8192, 128] f32
    float* out = (float*)d_out;               // [8192, 8192] f32

    dim3 grid(MDIM / TILE, NDIM / TILE);      // (64, 64)
    dim3 block(256);                          // 8 waves
    dist_wmma_f32_kernel<<<grid, block, 0, stream>>>(x, y, out);
}